// InnerProductDecoder2_73031623901827
// MI455X (gfx1250) — compile-verified
//
#include <hip/hip_runtime.h>

// CDNA5 / gfx1250. wave32. Uses V_WMMA_F32_16X16X4_F32:
//   wave handles 16 edges; A = 16 gathered src rows, B = 16 gathered dst rows^T,
//   diagonal of the 16x16 f32 accumulator = the 16 edge dot-products.
typedef __attribute__((ext_vector_type(2))) float v2f;
typedef __attribute__((ext_vector_type(8))) float v8f;

#define WAVES_PER_BLOCK 8
#define BLOCK_THREADS   (WAVES_PER_BLOCK * 32)
#define D_FEAT          64

__global__ __launch_bounds__(BLOCK_THREADS)
void InnerProductDecoder2_gumbel_edge_kernel(
    const float* __restrict__ z1,          // [N,64] f32
    const float* __restrict__ gu,          // [E,2]  f32
    const unsigned int* __restrict__ ei32, // edge_index viewed as u32; low word of int64 at 2*e
    float* __restrict__ out,               // [E]
    float* __restrict__ a_out,             // [E]
    int E, int numTiles)
{
    // per-wave 8x32 f32 scratch for diagonal extraction (1KB/wave, 8KB/block)
    __shared__ float diagLds[WAVES_PER_BLOCK * 8 * 32];

    const int lane        = threadIdx.x & 31;
    const int waveInBlock = threadIdx.x >> 5;
    const int waveGlobal  = blockIdx.x * WAVES_PER_BLOCK + waveInBlock;
    const int numWaves    = gridDim.x * WAVES_PER_BLOCK;
    const int h           = lane >> 4;   // 0: lanes 0-15 (K=0,1), 1: lanes 16-31 (K=2,3)
    const int m           = lane & 15;   // matrix row/col owned by this lane
    float* wlds = &diagLds[waveInBlock * 256];

    for (int tile = waveGlobal; tile < numTiles; tile += numWaves) {
        const int edge     = tile * 16 + m;
        const int eClamped = (edge < E) ? edge : (E - 1);    // branchless tail clamp

        // int64 indices: little-endian low 32 bits (N = 1e5 < 2^31) -> halve index traffic
        const int srcIdx = (int)ei32[2 * (size_t)eClamped];
        const int dstIdx = (int)ei32[2 * ((size_t)eClamped + (size_t)E)];

        // ISA A-layout (32-bit 16x4): lane m holds A[m][2h], A[m][2h+1] in v0,v1.
        // B-layout mirrors with N across lanes; both reduce to a contiguous float2
        // gather at rowBase + 4k + 2h. A uses src rows, B uses dst rows.
        const float* srcRow = z1 + (size_t)srcIdx * D_FEAT + 2 * h;
        const float* dstRow = z1 + (size_t)dstIdx * D_FEAT + 2 * h;

        v2f av[16], bv[16];
#pragma unroll
        for (int k = 0; k < 16; ++k) {                 // K = 64 total, 4 per step
            av[k] = *(const v2f*)(srcRow + 4 * k);     // 32 independent 8B gathers in flight
            bv[k] = *(const v2f*)(dstRow + 4 * k);
        }

        v8f c = {};
#pragma unroll
        for (int k = 0; k < 16; ++k) {
            c = __builtin_amdgcn_wmma_f32_16x16x4_f32(
                    /*neg_a=*/false, av[k], /*neg_b=*/false, bv[k],
                    /*c_mod=*/(short)0, c, /*reuse_a=*/false, /*reuse_b=*/false);
        }

        // C/D layout: VGPR r, lanes 0-15 -> (M=r, N=lane); lanes 16-31 -> (M=r+8, N=lane-16).
        // Diagonal (m,m): m<8 -> vgpr m @ lane m; m>=8 -> vgpr m-8 @ lane m+16.
        // Bounce through LDS (bank-conflict-free: 33*e mod 64 all distinct).
#pragma unroll
        for (int r = 0; r < 8; ++r)
            wlds[r * 32 + lane] = c[r];
        __builtin_amdgcn_wave_barrier();               // LDS ops are in-order within a wave

        const int rsel = (m < 8) ? m : (m - 8);
        const int csel = (m < 8) ? m : (m + 16);
        const float dotFull = wlds[rsel * 32 + csel];  // sum over ALL 64 dims

        if (h == 0 && edge < E) {                      // lanes 0-15: one edge each
            // correct to dims [2:64): subtract first-two-dim contribution
            const v2f zs = *(const v2f*)(z1 + (size_t)srcIdx * D_FEAT);
            const v2f zd = *(const v2f*)(z1 + (size_t)dstIdx * D_FEAT);
            const float vf = dotFull - (zs.x * zd.x + zs.y * zd.y);
            const float vn = zs.x + zd.x;

            // Gumbel: forward value of straight-through estimator is exactly `hard`;
            // softmax is monotone and temp>0 scales both logits -> temp drops out.
            const v2f u = *(const v2f*)(gu + 2 * (size_t)edge);
            const float eps = 1e-10f;
            const float u0 = fminf(fmaxf(u.x, eps), 1.0f - eps);
            const float u1 = fminf(fmaxf(u.y, eps), 1.0f - eps);
            const float g0 = -__logf(-__logf(u0));
            const float g1 = -__logf(-__logf(u1));
            const float flag = (vf + g0 >= g1) ? 1.0f : 0.0f;  // argmax ties -> index 0

            const float sf = 1.0f / (1.0f + __expf(-vf));
            const float sn = 1.0f / (1.0f + __expf(-vn));

            out[edge]   = flag * sf + (1.0f - flag) * sn;
            a_out[edge] = flag;
        }
    }
}

extern "C" void kernel_launch(void* const* d_in, const int* in_sizes, int n_in,
                              void* d_out, int out_size, void* d_ws, size_t ws_size,
                              hipStream_t stream) {
    // setup_inputs order: z1 [N*64 f32], gumbel_u [E*2 f32], temp [1], edge_index [2*E i64]
    const float*        z1   = (const float*)d_in[0];
    const float*        gu   = (const float*)d_in[1];
    const unsigned int* ei32 = (const unsigned int*)d_in[3];

    const int E = in_sizes[3] / 2;           // edge_index has 2*E int64 elements
    float* out   = (float*)d_out;            // outputs concatenated: out[E] then a[E]
    float* a_out = out + E;

    const int numTiles = (E + 15) / 16;      // 16 edges per wave-tile
    int blocks = (numTiles + WAVES_PER_BLOCK - 1) / WAVES_PER_BLOCK;
    if (blocks > 4096) blocks = 4096;        // grid-stride; ~6 tiles per wave at E=3.2M

    InnerProductDecoder2_gumbel_edge_kernel<<<blocks, BLOCK_THREADS, 0, stream>>>(
        z1, gu, ei32, out, a_out, E, numTiles);
}